// RoPE_3616362463612
// MI455X (gfx1250) — compile-verified
//
#include <hip/hip_runtime.h>

// RoPE, MI455X (gfx1250, wave32).
// x: (32, 8192, 128) f32, token_positions: (32, 8192) i32,
// sin_tab/cos_tab: (8192, 64) f32.  out: same shape as x.
//
// Bandwidth-bound (~270 MB HBM traffic, ~0.1 GFLOP -> AI ~0.37 FLOP/B).
// Strategy: one wave32 per row (128 floats = 32 lanes x float4), grid-stride
// persistent waves processing 2 rows/iter, NT hints on the streamed x/out,
// RT (default) hints on the 2 MB sin/cos tables so they stay L2-resident,
// and global_prefetch for the next iteration's x rows.

typedef __attribute__((ext_vector_type(4))) float v4f;
typedef __attribute__((ext_vector_type(2))) float v2f;

#define ROPE_BATCH   32
#define ROPE_SEQ     8192
#define ROPE_DK      128
#define ROPE_HALF    (ROPE_DK / 2)           // 64
#define ROPE_ROWS    (ROPE_BATCH * ROPE_SEQ) // 262144 (even)

__global__ __launch_bounds__(256) void rope_kernel(
    const float* __restrict__ x,
    const int*   __restrict__ tok_pos,
    const float* __restrict__ sin_tab,
    const float* __restrict__ cos_tab,
    float*       __restrict__ out,
    int numWaves)
{
    const int lane = threadIdx.x & 31;
    const int wave = blockIdx.x * (blockDim.x >> 5) + (threadIdx.x >> 5);

    // lane covers dim-pairs (2*lane, 2*lane+1): 16B of x, 8B of each table.
    const int xoff = lane * 4;   // float offset into 128-wide row
    const int toff = lane * 2;   // float offset into 64-wide table row

    const int rowStride = numWaves * 2;

    for (int row = wave * 2; row < ROPE_ROWS; row += rowStride) {
        const int row1 = row + 1;  // ROWS even, base even, stride even -> valid

        // Prefetch next iteration's x rows (gfx1250 global_prefetch_b8).
        const long nrow = (long)row + rowStride;
        if (nrow < ROPE_ROWS) {
            __builtin_prefetch(x + nrow       * (long)ROPE_DK + xoff, 0, 3);
            __builtin_prefetch(x + (nrow + 1) * (long)ROPE_DK + xoff, 0, 3);
        }

        // Wave-uniform positions -> SGPRs so table loads use saddr form.
        const int p0 = __builtin_amdgcn_readfirstlane(tok_pos[row]);
        const int p1 = __builtin_amdgcn_readfirstlane(tok_pos[row1]);

        // Streamed x: non-temporal b128 loads (read once).
        const v4f xv0 = __builtin_nontemporal_load(
            (const v4f*)(x + (long)row  * ROPE_DK + xoff));
        const v4f xv1 = __builtin_nontemporal_load(
            (const v4f*)(x + (long)row1 * ROPE_DK + xoff));

        // Tables: regular-temporal b64 loads, L2-resident (2 MB each).
        const v2f s0 = *(const v2f*)(sin_tab + (long)p0 * ROPE_HALF + toff);
        const v2f c0 = *(const v2f*)(cos_tab + (long)p0 * ROPE_HALF + toff);
        const v2f s1 = *(const v2f*)(sin_tab + (long)p1 * ROPE_HALF + toff);
        const v2f c1 = *(const v2f*)(cos_tab + (long)p1 * ROPE_HALF + toff);

        // Rotate pairs: (e,o) -> (c*e - s*o, s*e + c*o)
        v4f o0, o1;
        o0.x = __builtin_fmaf(c0.x, xv0.x, -s0.x * xv0.y);
        o0.y = __builtin_fmaf(s0.x, xv0.x,  c0.x * xv0.y);
        o0.z = __builtin_fmaf(c0.y, xv0.z, -s0.y * xv0.w);
        o0.w = __builtin_fmaf(s0.y, xv0.z,  c0.y * xv0.w);

        o1.x = __builtin_fmaf(c1.x, xv1.x, -s1.x * xv1.y);
        o1.y = __builtin_fmaf(s1.x, xv1.x,  c1.x * xv1.y);
        o1.z = __builtin_fmaf(c1.y, xv1.z, -s1.y * xv1.w);
        o1.w = __builtin_fmaf(s1.y, xv1.z,  c1.y * xv1.w);

        // Streamed out: non-temporal b128 stores (write once, never re-read).
        __builtin_nontemporal_store(o0, (v4f*)(out + (long)row  * ROPE_DK + xoff));
        __builtin_nontemporal_store(o1, (v4f*)(out + (long)row1 * ROPE_DK + xoff));
    }
}

extern "C" void kernel_launch(void* const* d_in, const int* in_sizes, int n_in,
                              void* d_out, int out_size, void* d_ws, size_t ws_size,
                              hipStream_t stream) {
    const float* x       = (const float*)d_in[0];
    const int*   tok_pos = (const int*)  d_in[1];
    const float* sin_tab = (const float*)d_in[2];
    const float* cos_tab = (const float*)d_in[3];
    float*       out     = (float*)d_out;

    // 4096 blocks x 256 threads = 32768 wave32s; 262144 rows -> each wave
    // loops 4 iterations of 2 rows (grid-stride), enabling the prefetch.
    const int blocks        = 4096;
    const int threads       = 256;
    const int wavesPerBlock = threads / 32;
    const int numWaves      = blocks * wavesPerBlock;

    rope_kernel<<<blocks, threads, 0, stream>>>(
        x, tok_pos, sin_tab, cos_tab, out, numWaves);
}